// GraphConvolution_21062519620169
// MI455X (gfx1250) — compile-verified
//
#include <hip/hip_runtime.h>

typedef __attribute__((ext_vector_type(2))) float v2f;
typedef __attribute__((ext_vector_type(4))) float v4f;
typedef __attribute__((ext_vector_type(8))) float v8f;

#define GC_N_NODES 100000
#define GC_N_EDGES 3200000
#define GC_IN_DIM 512
#define GC_OUT_DIM 128
#define GC_ROWS_PER_BLOCK 32                       // 2 M-tiles per wave
#define GC_ROW_BLOCKS (GC_N_NODES / GC_ROWS_PER_BLOCK)  // 3125 exact
#define GC_EDGES_PER_BLOCK 8                       // 8 waves per 256-thread block

// Hardware f32 atomic add, no return (L2-resolved). addr, data, off — same
// operand order as global_store_b32 in gfx1250 asm.
#define GC_ATOMIC_ADD_F32(P, V, OFF)                                     \
  asm volatile("global_atomic_add_f32 %0, %1, off offset:" #OFF          \
               :: "v"(P), "v"(V) : "memory")

// ---------------------------------------------------------------------------
// Kernel 0: zero the (poisoned) output accumulators.
// ---------------------------------------------------------------------------
__global__ __launch_bounds__(256) void gc_zero(float* __restrict__ out, int n4) {
  int i = blockIdx.x * blockDim.x + threadIdx.x;
  if (i < n4) {
    v4f z = {0.f, 0.f, 0.f, 0.f};
    ((v4f*)out)[i] = z;
  }
}

// ---------------------------------------------------------------------------
// Kernel 1: support = feature @ weight, exact f32 via V_WMMA_F32_16X16X4_F32.
// grid.x = 32-row block, grid.y = {ori, aug}.
// Block = 256 threads = 8 waves; wave w owns N-tile w (cols 16w..16w+15) and
// TWO 16-row M-tiles, sharing one B fragment per k-step between both WMMAs.
// All 8 waves share the same 32 feature rows (one HBM read, L2 broadcast);
// weight (256 KB) is L2-resident across the whole grid.
// ---------------------------------------------------------------------------
__global__ __launch_bounds__(256) void gc_gemm_wmma(
    const float* __restrict__ feat_ori,
    const float* __restrict__ feat_aug,
    const float* __restrict__ weight,    // [IN_DIM, OUT_DIM] row-major
    float* __restrict__ sup_ori,
    float* __restrict__ sup_aug) {
  const int rb   = blockIdx.x;
  const bool aug = (blockIdx.y != 0);
  const float* __restrict__ feat = aug ? feat_aug : feat_ori;
  float* __restrict__ sup        = aug ? sup_aug : sup_ori;

  const int wave = threadIdx.x >> 5;   // 0..7 -> N tile
  const int lane = threadIdx.x & 31;
  const int l    = lane & 15;          // M index (A) / N index (B, C/D)
  const int half = lane >> 4;          // 0: K=k,k+1   1: K=k+2,k+3
  const int n0   = wave * 16;
  const int row0 = rb * GC_ROWS_PER_BLOCK;

  // A fragments: lane reads feature[row + l][k + 2*half .. +1] as float2.
  const float* __restrict__ arow0 =
      feat + (size_t)(row0 + l) * GC_IN_DIM + 2 * half;
  const float* __restrict__ arow1 = arow0 + (size_t)16 * GC_IN_DIM;
  // B fragment: lane reads weight[k + 2*half][n0 + l], weight[k+1+2*half][n0+l].
  const float* __restrict__ bcol =
      weight + (size_t)(2 * half) * GC_OUT_DIM + n0 + l;

  v8f acc0 = {0.f, 0.f, 0.f, 0.f, 0.f, 0.f, 0.f, 0.f};
  v8f acc1 = {0.f, 0.f, 0.f, 0.f, 0.f, 0.f, 0.f, 0.f};

#pragma unroll 4
  for (int k = 0; k < GC_IN_DIM; k += 4) {
    v2f a0 = *(const v2f*)(arow0 + k);
    v2f a1 = *(const v2f*)(arow1 + k);
    v2f b;
    b.x = bcol[(size_t)k * GC_OUT_DIM];
    b.y = bcol[(size_t)(k + 1) * GC_OUT_DIM];
    acc0 = __builtin_amdgcn_wmma_f32_16x16x4_f32(
        false, a0, false, b, (short)0, acc0, false, false);
    acc1 = __builtin_amdgcn_wmma_f32_16x16x4_f32(
        false, a1, false, b, (short)0, acc1, false, false);
  }

  // C/D layout: VGPR r holds M = r (lanes 0-15) or M = r+8 (lanes 16-31).
  float* __restrict__ o0 =
      sup + (size_t)(row0 + 8 * half) * GC_OUT_DIM + n0 + l;
  float* __restrict__ o1 = o0 + (size_t)16 * GC_OUT_DIM;
#pragma unroll
  for (int r = 0; r < 8; ++r) {
    o0[(size_t)r * GC_OUT_DIM] = acc0[r];
    o1[(size_t)r * GC_OUT_DIM] = acc1[r];
  }
}

// ---------------------------------------------------------------------------
// Kernel 2: COO SpMM scatter. One wave per edge: coalesced 512B float4 gather
// of the support row (L2-resident), scale by edge_val, hardware f32 atomic
// scatter-add into the output rows (atomics resolve in L2).
// ---------------------------------------------------------------------------
__global__ __launch_bounds__(256) void gc_spmm_scatter(
    const int* __restrict__ erow,
    const int* __restrict__ ecol,
    const float* __restrict__ eval,
    const float* __restrict__ sup_ori,
    const float* __restrict__ sup_aug,
    float* __restrict__ out_ori,
    float* __restrict__ out_aug) {
  const int wave = threadIdx.x >> 5;
  const int lane = threadIdx.x & 31;
  const long long e = (long long)blockIdx.x * GC_EDGES_PER_BLOCK + wave;
  if (e >= GC_N_EDGES) return;

  const int r   = erow[e];
  const int c   = ecol[e];
  const float v = eval[e];

  const v4f* __restrict__ sa = (const v4f*)(sup_ori + (size_t)c * GC_OUT_DIM);
  const v4f* __restrict__ sb = (const v4f*)(sup_aug + (size_t)c * GC_OUT_DIM);
  v4f ma = sa[lane];   // 32 lanes x 16B = one coalesced 512B row
  v4f mb = sb[lane];
  ma *= v;
  mb *= v;

  float* __restrict__ oa = out_ori + (size_t)r * GC_OUT_DIM + lane * 4;
  float* __restrict__ ob = out_aug + (size_t)r * GC_OUT_DIM + lane * 4;
  GC_ATOMIC_ADD_F32(oa, ma.x, 0);
  GC_ATOMIC_ADD_F32(oa, ma.y, 4);
  GC_ATOMIC_ADD_F32(oa, ma.z, 8);
  GC_ATOMIC_ADD_F32(oa, ma.w, 12);
  GC_ATOMIC_ADD_F32(ob, mb.x, 0);
  GC_ATOMIC_ADD_F32(ob, mb.y, 4);
  GC_ATOMIC_ADD_F32(ob, mb.z, 8);
  GC_ATOMIC_ADD_F32(ob, mb.w, 12);
}

// ---------------------------------------------------------------------------
// Kernel 3: out = relu(out + bias), in place, float4-vectorized.
// ---------------------------------------------------------------------------
__global__ __launch_bounds__(256) void gc_bias_relu(
    float* __restrict__ out, const float* __restrict__ bias, int n4) {
  int i = blockIdx.x * blockDim.x + threadIdx.x;
  if (i >= n4) return;
  v4f v = ((v4f*)out)[i];
  const int c = (i * 4) & (GC_OUT_DIM - 1);   // column of this float4
  v4f b = *(const v4f*)(bias + c);
  v += b;
  v.x = fmaxf(v.x, 0.f);
  v.y = fmaxf(v.y, 0.f);
  v.z = fmaxf(v.z, 0.f);
  v.w = fmaxf(v.w, 0.f);
  ((v4f*)out)[i] = v;
}

// ---------------------------------------------------------------------------
// Host launcher. Inputs (setup_inputs order):
//   0 feature_ori [N,512] f32    1 feature_aug [N,512] f32
//   2 edge_row [E] i32           3 edge_col [E] i32
//   4 edge_val [E] f32           5 weight [512,128] f32   6 bias [128] f32
// d_out = output_ori [N,128] ++ output_aug [N,128] (f32).
// d_ws  = support_ori ++ support_aug (2 * 51.2 MB).
// ---------------------------------------------------------------------------
extern "C" void kernel_launch(void* const* d_in, const int* in_sizes, int n_in,
                              void* d_out, int out_size, void* d_ws, size_t ws_size,
                              hipStream_t stream) {
  const float* feat_ori = (const float*)d_in[0];
  const float* feat_aug = (const float*)d_in[1];
  const int*   erow     = (const int*)d_in[2];
  const int*   ecol     = (const int*)d_in[3];
  const float* eval     = (const float*)d_in[4];
  const float* weight   = (const float*)d_in[5];
  const float* bias     = (const float*)d_in[6];

  float* out_ori = (float*)d_out;
  float* out_aug = out_ori + (size_t)GC_N_NODES * GC_OUT_DIM;
  float* sup_ori = (float*)d_ws;
  float* sup_aug = sup_ori + (size_t)GC_N_NODES * GC_OUT_DIM;

  const int n4 = out_size / 4;   // 6.4M float4s

  // 1) zero the scatter accumulators (d_out is poisoned by the harness)
  gc_zero<<<(n4 + 255) / 256, 256, 0, stream>>>((float*)d_out, n4);

  // 2) support = feature @ weight, both matrices (grid.y)
  gc_gemm_wmma<<<dim3(GC_ROW_BLOCKS, 2), 256, 0, stream>>>(
      feat_ori, feat_aug, weight, sup_ori, sup_aug);

  // 3) edge scatter-add
  gc_spmm_scatter<<<(GC_N_EDGES + GC_EDGES_PER_BLOCK - 1) / GC_EDGES_PER_BLOCK,
                    256, 0, stream>>>(erow, ecol, eval, sup_ori, sup_aug,
                                      out_ori, out_aug);

  // 4) bias + relu in place
  gc_bias_relu<<<(n4 + 255) / 256, 256, 0, stream>>>((float*)d_out, bias, n4);
}